// Objective_Func_47596827574393
// MI455X (gfx1250) — compile-verified
//
#include <hip/hip_runtime.h>
#include <math.h>

// MI455X / gfx1250 implementation.
//
// ~5.1 GFLOP fp32 causal-conv scan (19 sequential steps x 1024 channels x
// 512x512 lower-triangular Toeplitz matvec), ~4.4 MB working set (L2-resident
// at 192MB) -> pure compute bound. Mapped onto V_WMMA_F32_16X16X4_F32
// (2048 FLOP/instr, full fp32), one wave32 per channel. Operands live in LDS
// with zero pads so the inner loop is branch-free; the conv kernel is stored
// un-reversed (q[u]) so A-operand pairs are ascending and load straight into
// even-aligned VGPR pairs with no fixup movs.

typedef __attribute__((ext_vector_type(2))) float v2f;
typedef __attribute__((ext_vector_type(8))) float v8f;

#define NLINKS 64
#define BATCH  16
#define NPOW   20
#define TLEN   512
#define NCHAN  (BATCH * NLINKS)   // 1024
#define DTf    0.01f
#define NOISEf 0.001f
#define LN2f   0.69314718055994531f
#define LOG2Ef 1.44269504088896340f

#define XPAD 256   // X column-block index down to -15 (16*-15 = -240 floats)
#define WTAIL 16   // q[] tail zeros: A pair base index up to 526

// ---------------------------------------------------------------------------
// sinr[b,n,l] = diag(PL)[l]*p / ((rowsum(PL)[l]-diag(PL)[l])*p + NOISE)
// ---------------------------------------------------------------------------
__global__ void sinr_kernel(const float* __restrict__ pathloss,
                            const float* __restrict__ powers,
                            float* __restrict__ sinr) {
    const int l = threadIdx.x;  // 0..63
    float rs = 0.0f;
    for (int j = 0; j < NLINKS; ++j) rs += pathloss[l * NLINKS + j];
    const float dg  = pathloss[l * NLINKS + l];
    const float off = rs - dg;
    for (int b = 0; b < BATCH; ++b) {
        for (int n = 0; n < NPOW; ++n) {
            const float p = powers[(b * NPOW + n) * NLINKS + l];
            sinr[(b * NPOW + n) * NLINKS + l] = dg * p / (off * p + NOISEf);
        }
    }
}

// ---------------------------------------------------------------------------
// Q1[b,l,t] = 1 - exp(-(2^(t*DT)-1) * sinr[b,0,l]);  outa[b,0,l] = Q1[...,T-1]
// ---------------------------------------------------------------------------
__global__ void q1_kernel(const float* __restrict__ sinr,
                          float* __restrict__ Q,
                          float* __restrict__ outa) {
    const int idx = blockIdx.x * blockDim.x + threadIdx.x;  // NCHAN*TLEN
    const int ch = idx >> 9;
    const int t  = idx & (TLEN - 1);
    const int b = ch >> 6, l = ch & 63;
    const float s    = sinr[b * (NPOW * NLINKS) + l];   // n = 0
    const float p2m1 = __builtin_amdgcn_exp2f((float)t * DTf) - 1.0f;
    const float q1   = 1.0f - __builtin_amdgcn_exp2f(-p2m1 * s * LOG2Ef);
    Q[idx] = q1;
    if (t == TLEN - 1) outa[b * (NPOW * NLINKS) + l] = q1;
}

// ---------------------------------------------------------------------------
// One scan step: Qn[t] = DT * sum_{m<=t} Qprev[t-m]*w[m], w[m] = q_n(T-1-m).
// Toeplitz-block GEMM: X[i,J] = Q[16J+i]; T_d[i,j] = w[16d+i-j] = q[511-16d-i+j]
//   => A pair for (d,kk,lane): {q[u0], q[u0+1]}, u0 = 511-16d-m+2kh+4kk
// Y[:, tile g] = sum_d T_d * X[:, (16g..16g+15)-d]
// ---------------------------------------------------------------------------
__global__ __launch_bounds__(256) void conv_step_kernel(
    const float* __restrict__ Qprev,
    float* __restrict__ Qnext,
    const float* __restrict__ sinr,
    float* __restrict__ outa,
    int n) {
    __shared__ __align__(16) float QsRaw[8][XPAD + TLEN];   // X, front pad
    __shared__ __align__(16) float WsRaw[8][TLEN + WTAIL];  // q, tail pad

    const int lane = threadIdx.x & 31;
    const int wid  = threadIdx.x >> 5;
    const int ch   = blockIdx.x * 8 + wid;   // 0..1023
    const int b = ch >> 6, l = ch & 63;
    const float s = sinr[b * (NPOW * NLINKS) + n * NLINKS + l];

    float* __restrict__ Xv = &QsRaw[wid][XPAD];
    float* __restrict__ Wq = &WsRaw[wid][0];

    // zero pads (per-wave region)
#pragma unroll
    for (int i = lane; i < XPAD; i += 32) QsRaw[wid][i] = 0.0f;
    if (lane < WTAIL) Wq[TLEN + lane] = 0.0f;

    // stage Q (b128) and build q[u] = exp(-(2^(u*DT)-1)s) * 2^(u*DT) * s * ln2
#pragma unroll
    for (int i = 0; i < 4; ++i) {
        const int m4 = 4 * (lane + 32 * i);
        *(float4*)(Xv + m4) = *(const float4*)(Qprev + ch * TLEN + m4);
        float4 w;
#pragma unroll
        for (int c = 0; c < 4; ++c) {
            const float tr = (float)(m4 + c) * DTf;
            const float p2 = __builtin_amdgcn_exp2f(tr);
            const float e  = __builtin_amdgcn_exp2f((1.0f - p2) * s * LOG2Ef);
            ((float*)&w)[c] = e * p2 * s * LN2f;
        }
        *(float4*)(Wq + m4) = w;
    }
    __syncthreads();

    const int m     = lane & 15;        // A row / B column for this lane
    const int kh    = lane >> 4;        // K-half (ISA 16x4 fp32 A/B layout)
    const int wbase = (TLEN - 1) - m + 2 * kh;

#pragma unroll
    for (int g = 0; g < 2; ++g) {
        v8f acc0 = {};
        v8f acc1 = {};
        const int dmax = 16 * g + 15;   // odd -> clean even/odd pairing

        auto gemm_d = [&](int d, v8f& acc) {
            const int cb = 16 * g + m - d;             // input block column
            const float* wp = Wq + (wbase - 16 * d);   // ascending q pairs
            const float* xp = Xv + 16 * cb + 2 * kh;   // even -> b64 loads
            const v2f A0 = {wp[0],  wp[1]};
            const v2f A1 = {wp[4],  wp[5]};
            const v2f A2 = {wp[8],  wp[9]};
            const v2f A3 = {wp[12], wp[13]};
            const v2f B0 = *(const v2f*)(xp);
            const v2f B1 = *(const v2f*)(xp + 4);
            const v2f B2 = *(const v2f*)(xp + 8);
            const v2f B3 = *(const v2f*)(xp + 12);
            acc = __builtin_amdgcn_wmma_f32_16x16x4_f32(false, A0, false, B0, (short)0, acc, false, false);
            acc = __builtin_amdgcn_wmma_f32_16x16x4_f32(false, A1, false, B1, (short)0, acc, false, false);
            acc = __builtin_amdgcn_wmma_f32_16x16x4_f32(false, A2, false, B2, (short)0, acc, false, false);
            acc = __builtin_amdgcn_wmma_f32_16x16x4_f32(false, A3, false, B3, (short)0, acc, false, false);
        };

#pragma unroll 4
        for (int d = 0; d <= dmax; d += 2) {
            gemm_d(d,     acc0);
            gemm_d(d + 1, acc1);
        }
        const v8f acc = acc0 + acc1;

        // C/D layout: VGPR r -> row (r+8*kh), col = m.  t = 16*colblk + 8*kh + r
        // => 8 consecutive t per lane: two b128 stores.
        float* outp = Qnext + ch * TLEN + 16 * (16 * g + m) + 8 * kh;
        float4 lo, hi;
        lo.x = DTf * acc[0]; lo.y = DTf * acc[1]; lo.z = DTf * acc[2]; lo.w = DTf * acc[3];
        hi.x = DTf * acc[4]; hi.y = DTf * acc[5]; hi.z = DTf * acc[6]; hi.w = DTf * acc[7];
        *(float4*)(outp)     = lo;
        *(float4*)(outp + 4) = hi;
        if (g == 1 && lane == 31) {  // holds t = 511 in acc[7]
            outa[b * (NPOW * NLINKS) + n * NLINKS + l] = DTf * acc[7];
        }
    }
}

// ---------------------------------------------------------------------------
// loss = sum_{b,l}[ p[b,0,l] + 1 + outa[b,19,l]
//                   + sum_{n=0..18} outa[b,n,l]*(p[b,n+1,l] + 1) ]
// ---------------------------------------------------------------------------
__global__ void loss_kernel(const float* __restrict__ powers,
                            const float* __restrict__ outa,
                            float* __restrict__ out) {
    __shared__ float red[256];
    float acc = 0.0f;
    for (int i = threadIdx.x; i < NCHAN; i += 256) {
        const int b = i >> 6, l = i & 63;
        const float* pw = powers + b * (NPOW * NLINKS) + l;
        const float* oa = outa   + b * (NPOW * NLINKS) + l;
        float v = pw[0] + 1.0f + oa[19 * NLINKS];
        for (int n = 0; n < 19; ++n)
            v += oa[n * NLINKS] * (pw[(n + 1) * NLINKS] + 1.0f);
        acc += v;
    }
    red[threadIdx.x] = acc;
    __syncthreads();
    for (int stride = 128; stride > 0; stride >>= 1) {
        if (threadIdx.x < (unsigned)stride) red[threadIdx.x] += red[threadIdx.x + stride];
        __syncthreads();
    }
    if (threadIdx.x == 0) out[0] = red[0];
}

// ---------------------------------------------------------------------------
// Workspace layout (floats):
//   sinr [20480] | outa [20480] | qping [524288] | qpong [524288]  (~4.4 MB)
// ---------------------------------------------------------------------------
extern "C" void kernel_launch(void* const* d_in, const int* in_sizes, int n_in,
                              void* d_out, int out_size, void* d_ws, size_t ws_size,
                              hipStream_t stream) {
    const float* pathloss = (const float*)d_in[0];
    const float* powers   = (const float*)d_in[1];
    float* out = (float*)d_out;
    float* ws  = (float*)d_ws;

    float* sinr  = ws;
    float* outa  = ws + 20480;
    float* qping = ws + 40960;
    float* qpong = qping + (size_t)NCHAN * TLEN;

    sinr_kernel<<<1, 64, 0, stream>>>(pathloss, powers, sinr);
    q1_kernel<<<(NCHAN * TLEN) / 256, 256, 0, stream>>>(sinr, qping, outa);

    float* qa = qping;
    float* qb = qpong;
    for (int k = 0; k < 19; ++k) {
        conv_step_kernel<<<NCHAN / 8, 256, 0, stream>>>(qa, qb, sinr, outa, k + 1);
        float* t = qa; qa = qb; qb = t;
    }

    loss_kernel<<<1, 256, 0, stream>>>(powers, outa, out);
}